// SAGEConvCuGraph_13048110645894
// MI455X (gfx1250) — compile-verified
//
#include <hip/hip_runtime.h>
#include <hip/hip_bf16.h>
#include <math.h>

// ---------------------------------------------------------------------------
// SAGEConv (mean aggregate) + Linear(concat) + L2 normalize, fp32 end-to-end.
// N=100000 nodes, E=1600000 edges, C_IN=64, C_OUT=64.
// ---------------------------------------------------------------------------

typedef __attribute__((ext_vector_type(2))) float v2f;
typedef __attribute__((ext_vector_type(8))) float v8f;

#define CIN   64
#define COUT  64
#define KDIM  128           // 2*CIN
#define SWP   132           // padded LDS pitch (128 + 4) to avoid bank conflicts

// ---------------------------------------------------------------------------
// Kernel 1: zero agg + deg (contiguous in workspace: N*64 + N floats)
// ---------------------------------------------------------------------------
__global__ void sage_init_kernel(float* __restrict__ ws, int total) {
    int i = blockIdx.x * blockDim.x + threadIdx.x;
    if (i < total) ws[i] = 0.0f;
}

// ---------------------------------------------------------------------------
// Kernel 2: edge scatter. 16 threads per edge, float4 per thread.
// agg[dst] += x[src]; deg[dst] += 1
// ---------------------------------------------------------------------------
__global__ void sage_scatter_kernel(const float* __restrict__ x,
                                    const long long* __restrict__ ei,
                                    float* __restrict__ agg,
                                    float* __restrict__ deg,
                                    int E) {
    int tid = blockIdx.x * blockDim.x + threadIdx.x;
    int e = tid >> 4;
    if (e >= E) return;
    int c4 = (tid & 15) << 2;                     // channel base 0..60
    long long src = ei[e];
    long long dst = ei[(long long)E + e];
    const float4 v = *reinterpret_cast<const float4*>(x + src * CIN + c4);
    float* a = agg + dst * CIN + c4;
    atomicAdd(a + 0, v.x);
    atomicAdd(a + 1, v.y);
    atomicAdd(a + 2, v.z);
    atomicAdd(a + 3, v.w);
    if ((tid & 15) == 0) atomicAdd(deg + dst, 1.0f);
}

// ---------------------------------------------------------------------------
// Kernel 3: WMMA f32 GEMM (h @ W^T + b) + row L2-normalize.
// Block = 256 threads = 8 waves; each wave owns a 16-row M-tile.
// LDS: sW [64 x SWP] + sH [8 waves][16 x SWP]  (dynamic shared)
// ---------------------------------------------------------------------------
__global__ void sage_gemm_norm_kernel(const float* __restrict__ x,
                                      const float* __restrict__ agg,
                                      const float* __restrict__ deg,
                                      const float* __restrict__ W,
                                      const float* __restrict__ bias,
                                      float* __restrict__ out,
                                      int N) {
    extern __shared__ float smem[];
    float* sW = smem;                 // 64 rows (n) x SWP (k)
    float* sH = smem + COUT * SWP;    // 8 * 16 rows (m) x SWP (k)

    const int tid  = threadIdx.x;
    const int wave = tid >> 5;
    const int lane = tid & 31;
    const int half = lane >> 4;       // 0 or 1
    const int l16  = lane & 15;

    // ---- stage W: 64 x 128 row-major -> padded LDS ----
    for (int i = tid; i < COUT * (KDIM / 4); i += 256) {
        int n  = i >> 5;              // 32 float4 per row
        int c4 = (i & 31) << 2;
        const float4 v = *reinterpret_cast<const float4*>(W + n * KDIM + c4);
        float* d = sW + n * SWP + c4;
        d[0] = v.x; d[1] = v.y; d[2] = v.z; d[3] = v.w;
    }

    // ---- stage this wave's 16-row h tile: [agg/deg | x] ----
    const int rowbase = blockIdx.x * 128 + wave * 16;
    float* myH = sH + wave * 16 * SWP;
    for (int r = 0; r < 16; ++r) {
        int row = rowbase + r;
        int c4  = lane << 2;                      // 0..124
        float* d = myH + r * SWP + c4;
        if (row < N) {
            if (lane < 16) {                      // aggregated mean half (k 0..63)
                float dg   = deg[row];
                float invd = 1.0f / fmaxf(dg, 1.0f);
                const float4 v = *reinterpret_cast<const float4*>(
                    agg + (size_t)row * CIN + c4);
                d[0] = v.x * invd; d[1] = v.y * invd;
                d[2] = v.z * invd; d[3] = v.w * invd;
            } else {                              // root half (k 64..127)
                const float4 v = *reinterpret_cast<const float4*>(
                    x + (size_t)row * CIN + (c4 - CIN));
                d[0] = v.x; d[1] = v.y; d[2] = v.z; d[3] = v.w;
            }
        } else {
            d[0] = 0.0f; d[1] = 0.0f; d[2] = 0.0f; d[3] = 0.0f;
        }
    }
    __syncthreads();

    // ---- 16x16x4 f32 WMMA over K=128, 4 N-tiles ----
    // A frag (16x4): lane M = l16; VGPR0 = K{0|2}, VGPR1 = K{1|3} by half.
    // B frag (4x16): lane N = l16; same K striping (mirror of A).
    v8f acc0 = {}, acc1 = {}, acc2 = {}, acc3 = {};
    const int koff = half * 2;
    const float* hrow = myH + l16 * SWP;
    const float* w0 = sW + (l16 + 0)  * SWP;
    const float* w1 = sW + (l16 + 16) * SWP;
    const float* w2 = sW + (l16 + 32) * SWP;
    const float* w3 = sW + (l16 + 48) * SWP;

    for (int ks = 0; ks < 32; ++ks) {
        int k = (ks << 2) + koff;
        v2f a, b0, b1, b2, b3;
        a.x  = hrow[k];  a.y  = hrow[k + 1];
        b0.x = w0[k];    b0.y = w0[k + 1];
        b1.x = w1[k];    b1.y = w1[k + 1];
        b2.x = w2[k];    b2.y = w2[k + 1];
        b3.x = w3[k];    b3.y = w3[k + 1];
        acc0 = __builtin_amdgcn_wmma_f32_16x16x4_f32(false, a, false, b0,
                                                     (short)0, acc0, false, false);
        acc1 = __builtin_amdgcn_wmma_f32_16x16x4_f32(false, a, false, b1,
                                                     (short)0, acc1, false, false);
        acc2 = __builtin_amdgcn_wmma_f32_16x16x4_f32(false, a, false, b2,
                                                     (short)0, acc2, false, false);
        acc3 = __builtin_amdgcn_wmma_f32_16x16x4_f32(false, a, false, b3,
                                                     (short)0, acc3, false, false);
    }

    // ---- epilogue: bias, row L2 norm, store ----
    // C/D layout: VGPR v, lane -> row = v + half*8, col = ntile*16 + l16
    float bv0 = bias[l16 +  0];
    float bv1 = bias[l16 + 16];
    float bv2 = bias[l16 + 32];
    float bv3 = bias[l16 + 48];

#pragma unroll
    for (int v = 0; v < 8; ++v) {
        float e0 = acc0[v] + bv0;
        float e1 = acc1[v] + bv1;
        float e2 = acc2[v] + bv2;
        float e3 = acc3[v] + bv3;
        float s = e0 * e0 + e1 * e1 + e2 * e2 + e3 * e3;
        // reduce across the 16 lanes of this half (one row per (v,half))
        s += __shfl_xor(s, 1);
        s += __shfl_xor(s, 2);
        s += __shfl_xor(s, 4);
        s += __shfl_xor(s, 8);
        float scale = 1.0f / fmaxf(sqrtf(s), 1e-12f);
        int row = rowbase + v + half * 8;
        if (row < N) {
            float* o = out + (size_t)row * COUT + l16;
            o[0]  = e0 * scale;
            o[16] = e1 * scale;
            o[32] = e2 * scale;
            o[48] = e3 * scale;
        }
    }
}

// ---------------------------------------------------------------------------
// Launch
// ---------------------------------------------------------------------------
extern "C" void kernel_launch(void* const* d_in, const int* in_sizes, int n_in,
                              void* d_out, int out_size, void* d_ws, size_t ws_size,
                              hipStream_t stream) {
    const float*     x    = (const float*)d_in[0];
    const long long* ei   = (const long long*)d_in[1];   // int64 [2, E]
    const float*     W    = (const float*)d_in[2];       // [64, 128]
    const float*     bias = (const float*)d_in[3];       // [64]
    float*           out  = (float*)d_out;

    const int N = in_sizes[0] / CIN;
    const int E = in_sizes[1] / 2;

    float* agg = (float*)d_ws;                 // N * 64
    float* deg = agg + (size_t)N * CIN;        // N

    // 1) zero accumulators
    int initElems = N * (CIN + 1);
    sage_init_kernel<<<(initElems + 255) / 256, 256, 0, stream>>>(agg, initElems);

    // 2) scatter-add messages
    long long sthreads = (long long)E * 16;
    sage_scatter_kernel<<<(int)((sthreads + 255) / 256), 256, 0, stream>>>(
        x, ei, agg, deg, E);

    // 3) WMMA GEMM + L2 normalize
    size_t shmem = (size_t)(COUT * SWP + 8 * 16 * SWP) * sizeof(float);
    sage_gemm_norm_kernel<<<(N + 127) / 128, 256, shmem, stream>>>(
        x, agg, deg, W, bias, out, N);
}